// AggregationGNN_DB_6777458393748
// MI455X (gfx1250) — compile-verified
//
#include <hip/hip_runtime.h>

// ---------------------------------------------------------------------------
// AggregationGNN forward on gfx1250 (MI455X)
//   - fp32 WMMA (v_wmma_f32_16x16x4_f32) for all matmuls
//   - Tensor Data Mover (tensor_load_to_lds) to stage S tiles / z row-blocks
// ---------------------------------------------------------------------------

typedef __attribute__((ext_vector_type(2))) float        v2f;
typedef __attribute__((ext_vector_type(8))) float        v8f;
typedef __attribute__((ext_vector_type(4))) unsigned int v4u;
typedef __attribute__((ext_vector_type(8))) int          v8i;
typedef __attribute__((ext_vector_type(4))) int          v4i;

#define BB   16
#define TT   64
#define F0c  8
#define NN   100
#define LK   16     // NEX+1 (conv sequence length)
#define F1c  32
#define F2c  64
#define DRD  128
#define HRD  64
#define OUTC 5
#define ZROW 128    // per-row z storage: F0c * LK

// D = A(16xK) * B(Kx16) + C via fp32 WMMA, K consumed 4 at a time.
// Fragment layouts (CDNA5 ISA 7.12.2, 32-bit, wave32):
//   A: lane L -> M = L%16 ; VGPR v -> K = 4*ks + 2*(L/16) + v
//   B: lane L -> N = L%16 ; VGPR v -> K = 4*ks + 2*(L/16) + v
//   C/D: lane L -> N = L%16 ; VGPR r -> M = r + 8*(L/16)
__device__ __forceinline__ v8f wmma4(v2f a, v2f b, v8f c) {
  return __builtin_amdgcn_wmma_f32_16x16x4_f32(
      /*neg_a=*/false, a, /*neg_b=*/false, b,
      /*c_mod=*/(short)0, c, /*reuse_a=*/false, /*reuse_b=*/false);
}

// ---------------------------------------------------------------------------
// TDM: load a 2-D fp32 tile (tile_d0 x tile_d1, contiguous dim0, line stride
// d0_stride elements) from global memory into LDS at lds_byte_off.
// D# layout per CDNA5 ISA (groups 0/1; groups 2/3 zero
// for <=2D). clang-23 / amdgpu-toolchain 6-arg builtin form.
// ---------------------------------------------------------------------------
__device__ __forceinline__ void tdm_load_2d_f32(unsigned lds_byte_off,
                                                const float* gsrc,
                                                unsigned tensor_d0, unsigned tensor_d1,
                                                unsigned tile_d0,   unsigned tile_d1,
                                                unsigned d0_stride) {
  unsigned long long ga = (unsigned long long)(const void*)gsrc;
  v4u g0;
  g0.x = 1u;                                          // count=1, user mode
  g0.y = lds_byte_off;                                // LDS byte address
  g0.z = (unsigned)(ga & 0xffffffffull);              // global_addr[31:0]
  g0.w = (unsigned)((ga >> 32) & 0x1ffffffull)        // global_addr[56:32]
         | (2u << 30);                                // type = 2 ("image")
  v8i g1;
  g1[0] = (int)(2u << 16);                            // data_size = 4 bytes
  g1[1] = (int)((tensor_d0 & 0xffffu) << 16);         // tensor_dim0[15:0]
  g1[2] = (int)((tensor_d0 >> 16) | ((tensor_d1 & 0xffffu) << 16));
  g1[3] = (int)((tensor_d1 >> 16) | ((tile_d0 & 0xffffu) << 16));
  g1[4] = (int)(tile_d1 & 0xffffu);                   // tile_dim1 ; tile_dim2=0
  g1[5] = (int)d0_stride;                             // tensor_dim0_stride[31:0]
  g1[6] = 0;
  g1[7] = 0;
  v4i g2 = {0, 0, 0, 0};
  v4i g3 = {0, 0, 0, 0};
  v8i g4 = {0, 0, 0, 0, 0, 0, 0, 0};                  // extra group (clang-23 form)
  __builtin_amdgcn_tensor_load_to_lds(g0, g1, g2, g3, g4, /*cpol=*/0);
}

__device__ __forceinline__ unsigned lds_off(const void* p) {
  return (unsigned)(unsigned long long)p;             // flat LDS: addr[31:0]
}

// ---------------------------------------------------------------------------
// K0: scatter x into z-layout:  Z2[(bt*100+n)*128 + f*16 + 0] = x[bt,f,n]
// ---------------------------------------------------------------------------
__global__ void __launch_bounds__(256)
scatter_x_kernel(const float* __restrict__ x, float* __restrict__ Z2) {
  int idx = blockIdx.x * blockDim.x + threadIdx.x;
  if (idx >= BB * TT * F0c * NN) return;
  int n  = idx % NN;
  int f  = (idx / NN) % F0c;
  int bt = idx / (NN * F0c);
  Z2[(size_t)(bt * NN + n) * ZROW + f * LK + 0] = x[idx];
}

// ---------------------------------------------------------------------------
// K1: one diffusion step:  z_k[b,t] = z_{k-1}[b,t-1] @ S[b,t]   (8x100x100)
//     grid = B*T blocks, 128 threads (4 waves), 7 N-tiles over the waves.
//     S[b,t] (40 KB) staged into LDS by the Tensor Data Mover.
// ---------------------------------------------------------------------------
__global__ void __launch_bounds__(128)
diffuse_step_kernel(const float* __restrict__ S, float* __restrict__ Z2, int k) {
  const int bt = blockIdx.x;
  const int t  = bt % TT;

  __shared__ float As[16 * 104];   // z_{k-1} rows f=0..15 (8..15 zero), n=0..99
  __shared__ float Ss[NN * NN];    // S[b,t] tile, 40000 B

  // TDM: DMA the full 100x100 S tile into LDS (wave 0 issues; per-wave cnt).
  if ((threadIdx.x >> 5) == 0) {
    tdm_load_2d_f32(lds_off(Ss), S + (size_t)bt * NN * NN,
                    /*tensor_d0=*/NN, /*tensor_d1=*/NN,
                    /*tile_d0=*/NN,   /*tile_d1=*/NN,
                    /*d0_stride=*/NN);
    __builtin_amdgcn_s_wait_tensorcnt(0);
  }

  // Stage z_{k-1}[b, t-1] into LDS (zero for t==0 / pad rows).
  for (int i = threadIdx.x; i < 16 * 104; i += 128) {
    int f = i / 104, n = i % 104;
    float v = 0.0f;
    if (t > 0 && f < F0c && n < NN)
      v = Z2[(size_t)((bt - 1) * NN + n) * ZROW + f * LK + (k - 1)];
    As[i] = v;
  }
  __syncthreads();

  const int lane = threadIdx.x & 31;
  const int wave = threadIdx.x >> 5;
  const int mloc = lane & 15;
  const int g    = lane >> 4;

  for (int nt = wave; nt < 7; nt += 4) {      // uniform per wave, no divergence
    const int ncol = nt * 16 + mloc;
    const int ncl  = ncol < NN ? ncol : NN - 1;   // clamp, stores guarded below
    v8f acc = {0.f, 0.f, 0.f, 0.f, 0.f, 0.f, 0.f, 0.f};
    for (int ks = 0; ks < 25; ++ks) {         // K = 100
      const int ka = ks * 4 + 2 * g;
      v2f a, b;
      a.x = As[mloc * 104 + ka];
      a.y = As[mloc * 104 + ka + 1];
      b.x = Ss[ka * NN + ncl];
      b.y = Ss[(ka + 1) * NN + ncl];
      acc = wmma4(a, b, acc);
    }
    if (ncol < NN && g == 0) {                // D rows M=0..7 live in g==0 lanes
      float* dst = &Z2[(size_t)(bt * NN + ncol) * ZROW + k];
#pragma unroll
      for (int r = 0; r < 8; ++r) dst[r * LK] = acc[r];  // f = r
    }
  }
}

// ---------------------------------------------------------------------------
// K2: fused head per 16-row tile (1 wave / block):
//     conv1(K=32)+relu+pool -> conv2(K=96)+relu+pool -> fc1(K=128)+relu -> fc2
//     z row-block (8 KB, contiguous) staged via TDM.
// ---------------------------------------------------------------------------
__global__ void __launch_bounds__(32)
head_kernel(const float* __restrict__ Z2,
            const float* __restrict__ w1,  const float* __restrict__ b1,
            const float* __restrict__ w2,  const float* __restrict__ b2,
            const float* __restrict__ fw1, const float* __restrict__ fb1,
            const float* __restrict__ fw2, const float* __restrict__ fb2,
            float* __restrict__ out) {
  const int tile = blockIdx.x;       // 16 consecutive rows m = tile*16 ..
  const int lane = threadIdx.x & 31;
  const int mloc = lane & 15;
  const int g    = lane >> 4;

  __shared__ float Zt[16 * ZROW];        // 2048 : z rows [f][k]
  __shared__ float Y1[16 * F1c * 6];     // 3072 : pooled conv1 [c1][lp]
  __shared__ float Y2[16 * DRD];         // 2048 : pooled conv2 [c2][lp2] (=fc in)
  __shared__ float Y3[16 * HRD];         // 1024 : fc1 out

  // TDM: contiguous 2048-float row block -> LDS (1-D transfer).
  tdm_load_2d_f32(lds_off(Zt), Z2 + (size_t)tile * 16 * ZROW,
                  /*tensor_d0=*/16 * ZROW, /*tensor_d1=*/1,
                  /*tile_d0=*/16 * ZROW,   /*tile_d1=*/1,
                  /*d0_stride=*/16 * ZROW);
  __builtin_amdgcn_s_wait_tensorcnt(0);
  __syncthreads();

  // ---- conv1: K = f*4+kk (32), N = c1 (32 -> 2 tiles), out l = 0..12, pool 2
  for (int lp = 0; lp < 6; ++lp) {
    for (int nt = 0; nt < 2; ++nt) {
      const int c1 = nt * 16 + mloc;
      v8f a0c = {0.f,0.f,0.f,0.f,0.f,0.f,0.f,0.f};
      v8f a1c = {0.f,0.f,0.f,0.f,0.f,0.f,0.f,0.f};
      for (int ks = 0; ks < 8; ++ks) {
        const int ka  = ks * 4 + 2 * g;
        const int fA  = ka >> 2,        kkA = ka & 3;
        const int fBk = (ka + 1) >> 2,  kkB = (ka + 1) & 3;
        v2f bf, a0, a1;
        bf.x = w1[c1 * 32 + ka];
        bf.y = w1[c1 * 32 + ka + 1];
        a0.x = Zt[mloc * ZROW + fA  * LK + (2 * lp)     + kkA];
        a0.y = Zt[mloc * ZROW + fBk * LK + (2 * lp)     + kkB];
        a1.x = Zt[mloc * ZROW + fA  * LK + (2 * lp + 1) + kkA];
        a1.y = Zt[mloc * ZROW + fBk * LK + (2 * lp + 1) + kkB];
        a0c = wmma4(a0, bf, a0c);
        a1c = wmma4(a1, bf, a1c);
      }
      const float bias = b1[c1];
#pragma unroll
      for (int r = 0; r < 8; ++r) {
        const int m = r + 8 * g;
        float v = fmaxf(a0c[r], a1c[r]) + bias;
        Y1[m * (F1c * 6) + c1 * 6 + lp] = fmaxf(v, 0.f);
      }
    }
  }
  __syncthreads();

  // ---- conv2: K = c1*3+kk (96), N = c2 (64 -> 4 tiles), out l = 0..3, pool 2
  for (int lp = 0; lp < 2; ++lp) {
    for (int nt = 0; nt < 4; ++nt) {
      const int c2 = nt * 16 + mloc;
      v8f a0c = {0.f,0.f,0.f,0.f,0.f,0.f,0.f,0.f};
      v8f a1c = {0.f,0.f,0.f,0.f,0.f,0.f,0.f,0.f};
      for (int ks = 0; ks < 24; ++ks) {
        const int ka = ks * 4 + 2 * g;
        const int cA = ka / 3,       kkA = ka % 3;
        const int cB = (ka + 1) / 3, kkB = (ka + 1) % 3;
        v2f bf, a0, a1;
        bf.x = w2[c2 * 96 + ka];
        bf.y = w2[c2 * 96 + ka + 1];
        a0.x = Y1[mloc * 192 + cA * 6 + (2 * lp)     + kkA];
        a0.y = Y1[mloc * 192 + cB * 6 + (2 * lp)     + kkB];
        a1.x = Y1[mloc * 192 + cA * 6 + (2 * lp + 1) + kkA];
        a1.y = Y1[mloc * 192 + cB * 6 + (2 * lp + 1) + kkB];
        a0c = wmma4(a0, bf, a0c);
        a1c = wmma4(a1, bf, a1c);
      }
      const float bias = b2[c2];
#pragma unroll
      for (int r = 0; r < 8; ++r) {
        const int m = r + 8 * g;
        float v = fmaxf(a0c[r], a1c[r]) + bias;
        Y2[m * DRD + c2 * 2 + lp] = fmaxf(v, 0.f);
      }
    }
  }
  __syncthreads();

  // ---- fc1: [16x128] x [128x64], relu
  for (int nt = 0; nt < 4; ++nt) {
    const int h = nt * 16 + mloc;
    v8f acc = {0.f,0.f,0.f,0.f,0.f,0.f,0.f,0.f};
    for (int ks = 0; ks < 32; ++ks) {
      const int ka = ks * 4 + 2 * g;
      v2f a, bf;
      a.x  = Y2[mloc * DRD + ka];
      a.y  = Y2[mloc * DRD + ka + 1];
      bf.x = fw1[h * DRD + ka];
      bf.y = fw1[h * DRD + ka + 1];
      acc = wmma4(a, bf, acc);
    }
    const float bias = fb1[h];
#pragma unroll
    for (int r = 0; r < 8; ++r) {
      const int m = r + 8 * g;
      Y3[m * HRD + h] = fmaxf(acc[r] + bias, 0.f);
    }
  }
  __syncthreads();

  // ---- fc2: [16x64] x [64x5(pad16)] ; write out[b,t,o,n]
  {
    const int o  = mloc;                       // valid for o < 5
    const int oc = o < OUTC ? o : OUTC - 1;    // branchless clamp
    const float msk = o < OUTC ? 1.f : 0.f;
    v8f acc = {0.f,0.f,0.f,0.f,0.f,0.f,0.f,0.f};
    for (int ks = 0; ks < 16; ++ks) {
      const int ka = ks * 4 + 2 * g;
      v2f a, bf;
      a.x  = Y3[mloc * HRD + ka];
      a.y  = Y3[mloc * HRD + ka + 1];
      bf.x = fw2[oc * HRD + ka] * msk;
      bf.y = fw2[oc * HRD + ka + 1] * msk;
      acc = wmma4(a, bf, acc);
    }
    if (o < OUTC) {
      const float bias = fb2[o];
#pragma unroll
      for (int r = 0; r < 8; ++r) {
        const int m  = tile * 16 + r + 8 * g;
        const int bt = m / NN, n = m % NN;
        out[(size_t)bt * (OUTC * NN) + o * NN + n] = acc[r] + bias;
      }
    }
  }
}

// ---------------------------------------------------------------------------
extern "C" void kernel_launch(void* const* d_in, const int* in_sizes, int n_in,
                              void* d_out, int out_size, void* d_ws, size_t ws_size,
                              hipStream_t stream) {
  (void)in_sizes; (void)n_in; (void)out_size; (void)ws_size;
  const float* x   = (const float*)d_in[0];
  const float* S   = (const float*)d_in[1];
  const float* w1  = (const float*)d_in[2];
  const float* b1  = (const float*)d_in[3];
  const float* w2  = (const float*)d_in[4];
  const float* b2  = (const float*)d_in[5];
  const float* fw1 = (const float*)d_in[6];
  const float* fb1 = (const float*)d_in[7];
  const float* fw2 = (const float*)d_in[8];
  const float* fb2 = (const float*)d_in[9];
  float* out = (float*)d_out;
  float* Z2  = (float*)d_ws;     // [B*T*N][F0][LK] fp32 = 52.4 MB

  // z_0 = x
  scatter_x_kernel<<<(BB * TT * F0c * NN + 255) / 256, 256, 0, stream>>>(x, Z2);

  // z_k = z_{k-1}[t-1] @ S[t]  (serial over k, parallel over (b,t))
  for (int k = 1; k <= 15; ++k)
    diffuse_step_kernel<<<BB * TT, 128, 0, stream>>>(S, Z2, k);

  // conv1 -> pool -> conv2 -> pool -> fc1 -> fc2, 16 rows per wave
  head_kernel<<<(BB * TT * NN) / 16, 32, 0, stream>>>(
      Z2, w1, b1, w2, b2, fw1, fb1, fw2, fb2, out);
}